// SimpleGCN_15745350107435
// MI455X (gfx1250) — compile-verified
//
#include <hip/hip_runtime.h>
#include <stdint.h>

#define N_NODES 50000
#define N_EDGES 800000
#define D 128

#define NWAVES 8      // waves per workgroup in the fused MLP kernel
#define HLD 132       // padded LDS stride (floats) for the h staging tile

typedef float v2f __attribute__((ext_vector_type(2)));
typedef float v8f __attribute__((ext_vector_type(8)));

// ---------------------------------------------------------------------------
// Kernel 1: initialize x2 (segment-max accumulator) to -inf bit pattern.
// N_NODES*D = 6,400,000 floats = 1,600,000 uint4 stores.
// ---------------------------------------------------------------------------
__global__ __launch_bounds__(256) void init_x2(uint32_t* __restrict__ x2) {
  size_t i = ((size_t)blockIdx.x * blockDim.x + threadIdx.x) * 4;
  if (i < (size_t)N_NODES * D) {
    uint4 v = make_uint4(0xFF800000u, 0xFF800000u, 0xFF800000u, 0xFF800000u);
    *(uint4*)(x2 + i) = v;
  }
}

// ---------------------------------------------------------------------------
// Kernel 2: edge gather + scatter-max. 128 lanes handle one edge's 128
// features (coalesced 512B row read). Float max via sign-split int atomics:
//   v >= +0 : atomicMax on int   (int order == float order for non-neg)
//   v <  -0 : atomicMin on uint  (uint order reversed for negatives)
// Mixed signs interact correctly because any non-negative bit pattern
// (< 0x80000000) beats any negative one under both orderings.
// ---------------------------------------------------------------------------
__global__ __launch_bounds__(256) void scatter_max(const float* __restrict__ x1,
                                                   const long long* __restrict__ adj,
                                                   float* __restrict__ x2) {
  int e = blockIdx.x * 2 + (threadIdx.x >> 7);
  int d = threadIdx.x & 127;
  if (e >= N_EDGES) return;
  int src = (int)adj[e];            // adj[0][e]
  int dst = (int)adj[N_EDGES + e];  // adj[1][e]
  float v = x1[(size_t)src * D + d];
  float* addr = x2 + (size_t)dst * D + d;
  unsigned int bits = __float_as_uint(v);
  if (bits < 0x80000000u) {
    atomicMax((int*)addr, (int)bits);
  } else {
    atomicMin((unsigned int*)addr, bits);
  }
}

// ---------------------------------------------------------------------------
// Kernel 3: fused  out = relu((x1 + fix(x2)) @ W1 + b1) @ W2 + b2
// One wave per 16-row tile; V_WMMA_F32_16X16X4_F32 (exact fp32).
// Weights staged in LDS pre-swizzled into B-fragment order so each fragment
// is one conflict-free ds_load_b64. Same 64KB LDS buffer reused for W1/W2
// with a workgroup barrier between layers. Inter-layer transpose goes
// through a padded per-wave LDS staging tile.
//
// Fragment layouts (ISA 7.12.2):
//   A 16x4 f32, 2 VGPRs: lanes 0-15 row m, K={k0,k0+1}; lanes 16-31 K={k0+2,k0+3}
//   B 4x16 f32, 2 VGPRs: v0 = row k0 (lanes 0-15) / row k0+2 (lanes 16-31); v1 = +1
//   C/D 16x16 f32, 8 VGPRs: vj -> M=j (lanes 0-15), M=8+j (lanes 16-31), N=lane&15
// ---------------------------------------------------------------------------
__global__ __launch_bounds__(256) void fused_mlp(
    const float* __restrict__ x1, const float* __restrict__ x2,
    const float* __restrict__ W1, const float* __restrict__ b1,
    const float* __restrict__ W2, const float* __restrict__ b2,
    float* __restrict__ out) {
  extern __shared__ float lds[];
  float* ldsW  = lds;               // 128*128 floats, fragment-swizzled weights
  float* stage = lds + D * D;       // NWAVES * 16 * HLD floats, h staging

  const int tid  = threadIdx.x;
  const int wave = tid >> 5;
  const int lane = tid & 31;
  const int m    = lane & 15;        // output column within 16-wide tile / A row
  const int koff = (lane >> 4) << 1; // 0 (lanes 0-15) or 2 (lanes 16-31)
  const int half = lane >> 4;

  float* myStage = stage + wave * (16 * HLD);

  const int  tile   = blockIdx.x * NWAVES + wave; // 16-row tile index
  const bool active = tile < (N_NODES / 16);      // 3125 tiles exactly
  const int  row0   = tile * 16;

  // ---- stage W1 into LDS in WMMA-B fragment order ----
  // memory (r=K, c=N) -> fragment slot:
  //   kc=r>>2, kr=r&3, n=c>>4, mm=c&15
  //   lane = mm + 16*(kr>=2), vgpr-slot = kr&1
  for (int i = tid; i < D * D; i += 256) {
    int r = i >> 7, c = i & 127;
    int kc = r >> 2, kr = r & 3;
    int n = c >> 4, mm = c & 15;
    int l2 = ((kr & 2) << 3) | mm;
    ldsW[((((kc << 3) | n) << 5) + l2) * 2 + (kr & 1)] = W1[i];
  }
  __syncthreads();

  v8f acc[8];
  if (active) {
    #pragma unroll
    for (int n = 0; n < 8; ++n)
      #pragma unroll
      for (int j = 0; j < 8; ++j) acc[n][j] = 0.0f;

    const float* r1 = x1 + (size_t)(row0 + m) * D;
    const float* r2 = x2 + (size_t)(row0 + m) * D;
    for (int kc = 0; kc < 32; ++kc) {
      const int k0 = (kc << 2) + koff;
      float2 u = *(const float2*)(r1 + k0);
      float2 w = *(const float2*)(r2 + k0);
      v2f a;
      a[0] = u.x + ((__float_as_uint(w.x) == 0xFF800000u) ? 0.0f : w.x);
      a[1] = u.y + ((__float_as_uint(w.y) == 0xFF800000u) ? 0.0f : w.y);
      #pragma unroll
      for (int n = 0; n < 8; ++n) {
        v2f bf = *(const v2f*)&ldsW[((((kc << 3) | n) << 5) + lane) * 2];
        acc[n] = __builtin_amdgcn_wmma_f32_16x16x4_f32(
            false, a, false, bf, (short)0, acc[n], false, false);
      }
    }
    // bias + relu, stash h (C layout -> row-major staging tile)
    #pragma unroll
    for (int n = 0; n < 8; ++n) {
      float bv = b1[(n << 4) | m];
      #pragma unroll
      for (int j = 0; j < 8; ++j) {
        float h = acc[n][j] + bv;
        h = h > 0.0f ? h : 0.0f;
        myStage[(j + (half << 3)) * HLD + (n << 4) + m] = h;
      }
    }
  }
  __syncthreads();

  // ---- overwrite LDS with W2 fragments ----
  for (int i = tid; i < D * D; i += 256) {
    int r = i >> 7, c = i & 127;
    int kc = r >> 2, kr = r & 3;
    int n = c >> 4, mm = c & 15;
    int l2 = ((kr & 2) << 3) | mm;
    ldsW[((((kc << 3) | n) << 5) + l2) * 2 + (kr & 1)] = W2[i];
  }
  __syncthreads();

  if (active) {
    #pragma unroll
    for (int n = 0; n < 8; ++n)
      #pragma unroll
      for (int j = 0; j < 8; ++j) acc[n][j] = 0.0f;

    const float* hrow = myStage + m * HLD;
    for (int kc = 0; kc < 32; ++kc) {
      const int k0 = (kc << 2) + koff;
      v2f a = *(const v2f*)(hrow + k0);
      #pragma unroll
      for (int n = 0; n < 8; ++n) {
        v2f bf = *(const v2f*)&ldsW[((((kc << 3) | n) << 5) + lane) * 2];
        acc[n] = __builtin_amdgcn_wmma_f32_16x16x4_f32(
            false, a, false, bf, (short)0, acc[n], false, false);
      }
    }
    #pragma unroll
    for (int n = 0; n < 8; ++n) {
      float bv = b2[(n << 4) | m];
      #pragma unroll
      for (int j = 0; j < 8; ++j) {
        out[(size_t)(row0 + j + (half << 3)) * D + (n << 4) + m] = acc[n][j] + bv;
      }
    }
  }
}

// ---------------------------------------------------------------------------
extern "C" void kernel_launch(void* const* d_in, const int* in_sizes, int n_in,
                              void* d_out, int out_size, void* d_ws, size_t ws_size,
                              hipStream_t stream) {
  (void)in_sizes; (void)n_in; (void)out_size; (void)ws_size;
  const float*     x1  = (const float*)d_in[0];
  const long long* adj = (const long long*)d_in[1];  // int64 [2, E]
  const float*     W1  = (const float*)d_in[2];
  const float*     b1  = (const float*)d_in[3];
  const float*     W2  = (const float*)d_in[4];
  const float*     b2  = (const float*)d_in[5];
  float* out = (float*)d_out;
  float* x2  = (float*)d_ws;  // N_NODES*D floats = 25.6 MB scratch

  // 1) x2 = -inf
  int initGrid = (N_NODES * D + 256 * 4 - 1) / (256 * 4); // 6250
  init_x2<<<initGrid, 256, 0, stream>>>((uint32_t*)x2);

  // 2) scatter-max (2 edges per 256-thread block)
  scatter_max<<<(N_EDGES + 1) / 2, 256, 0, stream>>>(x1, adj, x2);

  // 3) fused residual + 2-layer MLP with fp32 WMMA
  size_t shmem = (size_t)(D * D + NWAVES * 16 * HLD) * sizeof(float); // ~130 KB
  int ntiles = N_NODES / 16;                       // 3125
  int grid = (ntiles + NWAVES - 1) / NWAVES;       // 391
  fused_mlp<<<grid, 256, shmem, stream>>>(x1, x2, W1, b1, W2, b2, out);
}